// FullLatticeFieldAnalyzer_77300821393855
// MI455X (gfx1250) — compile-verified
//
#include <hip/hip_runtime.h>
#include <hip/hip_bf16.h>
#include <math.h>

typedef __attribute__((ext_vector_type(2))) float v2f;
typedef __attribute__((ext_vector_type(8))) float v8f;

#define S_DIM 8192
#define D_DIM 1024

// Fully fused: every block recomputes the tiny anchor-weight MLP (256 FLOPs)
// in thread 0 -> LDS, then all 8 waves do the gather-einsum via
// V_WMMA_F32_16X16X4_F32 (one 16-wide d-tile per wave).
//
//   out[b,d] = iw[A-1][d] * sum_a w[a] * hs[b, idx[a], d]
//
// GEMM mapping (M=16 d-values, K=4 anchors, N=16 redundant columns):
//   D[m,n] = sum_k A[m,k]*B[k,n],  A[m,k] = hs[b, idx[k], d0+m],  B[k,n] = w[k]
__global__ void __launch_bounds__(256)
lattice_fused_wmma_kernel(const float* __restrict__ hs,
                          const float* __restrict__ iw,
                          const float* __restrict__ w1,
                          const float* __restrict__ b1,
                          const float* __restrict__ w2,
                          const float* __restrict__ b2,
                          const int*   __restrict__ position_p,
                          float* __restrict__ out) {
    __shared__ float s_w[4];
    __shared__ int   s_idx[4];
    __shared__ int   s_arow;

    if (threadIdx.x == 0) {
        const int position = position_p[0];

        // Only spine values < 13 can form path-count keys (sp, position):
        // candidates {0,1,2,5}; count = fib-like f(position-sp), f(1)=1, f(2)=2.
        const int sp[4] = {0, 1, 2, 5};
        int   idx[4] = {0, 0, 0, 0};
        float cnt[4] = {0.f, 0.f, 0.f, 0.f};
        int   A = 0;

        if (position >= 1 && position <= 12) {
            for (int k = 0; k < 4; ++k) {
                int s = sp[k];
                if (s < position && s < S_DIM) {
                    int d = position - s;            // >= 1
                    float f1 = 1.f, f2 = 2.f;
                    float f  = (d == 1) ? 1.f : 2.f;
                    for (int t = 3; t <= d; ++t) { f = f2 + f1; f1 = f2; f2 = f; }
                    idx[A] = s;
                    cnt[A] = f;
                    ++A;
                }
            }
        }

        // h = relu(c*w1 + b1) (A x 64); raw = softplus(h . w2 + b2); normalize.
        float sraw[4] = {0.f, 0.f, 0.f, 0.f};
        float ssum = 0.f;
        for (int a = 0; a < A; ++a) {
            float acc = 0.f;
            for (int j = 0; j < 64; ++j) {
                float h = fmaf(cnt[a], w1[j], b1[j]);
                h = fmaxf(h, 0.f);
                acc = fmaf(h, w2[j], acc);
            }
            float z = acc + b2[0];
            float spv = (z > 20.f) ? z : log1pf(__expf(z));   // stable softplus
            sraw[a] = spv;
            ssum += spv;
        }

        for (int a = 0; a < 4; ++a)
            s_w[a] = (a < A && ssum != 0.f) ? (sraw[a] / ssum) : 0.f;
        s_idx[0] = idx[0]; s_idx[1] = idx[1]; s_idx[2] = idx[2]; s_idx[3] = idx[3];
        s_arow = (A > 0) ? (A - 1) : 0;
    }
    __syncthreads();   // all lanes reconverge -> EXEC all-1s for the WMMA

    const int wave = threadIdx.x >> 5;              // 8 waves / block
    const int lane = threadIdx.x & 31;
    const int tile = blockIdx.x * 8 + wave;         // 0..511
    const int b    = tile >> 6;                     // tile / (D/16)
    const int d0   = (tile & 63) << 4;              // 16-wide d tile

    const int m    = lane & 15;                     // M = row within tile
    const int half = lane >> 4;                     // 0: K=0,1   1: K=2,3

    // A operand: 16x4 f32 — lanes 0-15: VGPR0=K0, VGPR1=K1; lanes 16-31: K2,K3
    const int rA = half ? s_idx[2] : s_idx[0];
    const int rB = half ? s_idx[3] : s_idx[1];
    const size_t base = (size_t)b * S_DIM * D_DIM + (size_t)(d0 + m);
    v2f amat;
    amat.x = hs[base + (size_t)rA * D_DIM];
    amat.y = hs[base + (size_t)rB * D_DIM];

    // B operand: 4x16 f32 — same K striping; all N columns get w[k] (broadcast)
    v2f bmat;
    bmat.x = half ? s_w[2] : s_w[0];
    bmat.y = half ? s_w[3] : s_w[1];

    v8f c = {};
    // 8 args: (neg_a, A, neg_b, B, c_mod, C, reuse_a, reuse_b)
    c = __builtin_amdgcn_wmma_f32_16x16x4_f32(false, amat, false, bmat,
                                              (short)0, c, false, false);

    // D layout: VGPR r -> lanes 0-15: (M=r, N=lane); lanes 16-31: (M=r+8, N=lane-16).
    // All columns identical: lane 0 holds Y[d0+0..7] in c[0..7], lane 16 holds Y[d0+8..15].
    if (m == 0) {
        const int dbase = d0 + half * 8;
        const float* iwrow = iw + (size_t)s_arow * D_DIM + dbase;
        float* o = out + (size_t)b * D_DIM + dbase;
#pragma unroll
        for (int r = 0; r < 8; ++r)
            o[r] = c[r] * iwrow[r];
    }
}

extern "C" void kernel_launch(void* const* d_in, const int* in_sizes, int n_in,
                              void* d_out, int out_size, void* d_ws, size_t ws_size,
                              hipStream_t stream) {
    const float* hs  = (const float*)d_in[0];   // (8, 8192, 1024) f32
    const float* iw  = (const float*)d_in[1];   // (10, 1024) f32
    const float* w1  = (const float*)d_in[2];   // (64,1)
    const float* b1  = (const float*)d_in[3];   // (64,)
    const float* w2  = (const float*)d_in[4];   // (1,64)
    const float* b2  = (const float*)d_in[5];   // (1,)
    const int*   pos = (const int*)d_in[6];     // scalar
    float* o = (float*)d_out;                   // (8, 1024) f32

    // 512 tiles of 16 outputs = 8*1024; 8 waves/block -> 64 blocks, single launch.
    lattice_fused_wmma_kernel<<<64, 256, 0, stream>>>(hs, iw, w1, b1, w2, b2, pos, o);
}